// GraphTransformerLinkPredictor_76003741270523
// MI455X (gfx1250) — compile-verified
//
#include <hip/hip_runtime.h>
#include <hip/hip_bf16.h>
#include <math.h>

#define HID 128
#define HEADS 4
#define DH 32
#define NLAYERS 2
#define RWSE_DIM 16
#define WALK_LEN 16
#define IN_C 128
#define LN_EPS 1e-5f

typedef __attribute__((ext_vector_type(2))) float v2f;
typedef __attribute__((ext_vector_type(8))) float v8f;

__device__ __forceinline__ float wave_red_sum(float v) {
  v += __shfl_xor(v, 16, 32);
  v += __shfl_xor(v, 8, 32);
  v += __shfl_xor(v, 4, 32);
  v += __shfl_xor(v, 2, 32);
  v += __shfl_xor(v, 1, 32);
  return v;
}

// Sum across the 8 lanes sharing lane>>3 (one head's lanes).
__device__ __forceinline__ float head_red_sum(float v) {
  v += __shfl_xor(v, 4, 32);
  v += __shfl_xor(v, 2, 32);
  v += __shfl_xor(v, 1, 32);
  return v;
}

// Order-preserving float -> uint mapping for atomicMax-based segment max.
__device__ __forceinline__ unsigned fkey(float f) {
  unsigned u = __float_as_uint(f);
  return (u & 0x80000000u) ? ~u : (u | 0x80000000u);
}
__device__ __forceinline__ float funkey(unsigned u) {
  return (u & 0x80000000u) ? __uint_as_float(u & 0x7FFFFFFFu)
                           : __uint_as_float(~u);
}

__device__ __forceinline__ void atomic_fadd(float* p, float v) {
  __hip_atomic_fetch_add(p, v, __ATOMIC_RELAXED, __HIP_MEMORY_SCOPE_AGENT);
}

// pe[n][j] = sum_k rw_diag[n][k] * W_rwse[k][j] + b_rwse[j]
__global__ void pe_kernel(const float* __restrict__ rw, const float* __restrict__ W,
                          const float* __restrict__ b, float* __restrict__ pe, int N) {
  int idx = blockIdx.x * blockDim.x + threadIdx.x;
  int n = idx >> 4;
  int j = idx & 15;
  if (n >= N) return;
  float acc = b[j];
  const float* r = rw + (size_t)n * WALK_LEN;
#pragma unroll
  for (int k = 0; k < WALK_LEN; ++k) acc += r[k] * W[k * RWSE_DIM + j];
  pe[(size_t)n * RWSE_DIM + j] = acc;
}

// Repack W[K x 128] into per-lane B fragments (fragment order):
// Wp[((ks*8)+nt)*32 + lane] = { W[ka][col], W[ka+1][col] },
//   ka = 4*ks + 2*(lane>=16), col = nt*16 + (lane&15).
__global__ void pack_w_kernel(const float* __restrict__ W, float2* __restrict__ Wp, int K) {
  int idx = blockIdx.x * blockDim.x + threadIdx.x;
  int total = (K >> 2) << 8;  // (K/4) * 8 * 32
  if (idx >= total) return;
  int lane = idx & 31;
  int nt = (idx >> 5) & 7;
  int ks = idx >> 8;
  int ka = (ks << 2) + ((lane >> 4) << 1);
  int col = nt * 16 + (lane & 15);
  Wp[idx] = make_float2(W[(size_t)ka * HID + col], W[(size_t)(ka + 1) * HID + col]);
}

// C[nrows x 128] = A[nrows x K] (lda) @ W (pre-packed fragments) (+ bias | + C).
// One wave computes a 16x128 stripe via V_WMMA_F32_16X16X4_F32.
__global__ void __launch_bounds__(128)
gemm128_wmma(const float* __restrict__ A, int lda, int K,
             const float2* __restrict__ Wp,
             const float* __restrict__ bias,
             float* __restrict__ C, int nrows, int accumulate) {
  int wavesPerBlock = blockDim.x >> 5;
  int rowTile = blockIdx.x * wavesPerBlock + (threadIdx.x >> 5);
  int nTiles = nrows >> 4;
  if (rowTile >= nTiles) return;
  int lane = threadIdx.x & 31;
  int hi = lane >> 4;
  int l16 = lane & 15;
  int m0 = rowTile << 4;
  const float* Arow = A + (size_t)(m0 + l16) * lda;

  v8f acc[8];
  int r0 = m0 + (hi ? 8 : 0);   // C/D: reg r -> row m0 + r + 8*hi, col = nt*16 + l16
  if (accumulate) {
#pragma unroll
    for (int nt = 0; nt < 8; ++nt)
#pragma unroll
      for (int r = 0; r < 8; ++r)
        acc[nt][r] = C[(size_t)(r0 + r) * HID + nt * 16 + l16];
  } else {
#pragma unroll
    for (int nt = 0; nt < 8; ++nt) {
      float bv = bias ? bias[nt * 16 + l16] : 0.0f;
#pragma unroll
      for (int r = 0; r < 8; ++r) acc[nt][r] = bv;
    }
  }

  int ksteps = K >> 2;
  const float* ap = Arow + (hi << 1);
  const float2* wp = Wp + lane;
  for (int ks = 0; ks < ksteps; ++ks) {
    // Stage all fragments for this k-step first so the loads clause and the
    // waits stagger (instead of one load register forcing wait-0 per WMMA).
    float2 af = *(const float2*)ap;
    float2 bf[8];
#pragma unroll
    for (int nt = 0; nt < 8; ++nt) bf[nt] = wp[nt << 5];
    v2f a;
    a.x = af.x;
    a.y = af.y;
#pragma unroll
    for (int nt = 0; nt < 8; ++nt) {
      v2f b;
      b.x = bf[nt].x;
      b.y = bf[nt].y;
      acc[nt] = __builtin_amdgcn_wmma_f32_16x16x4_f32(
          false, a, false, b, (short)0, acc[nt], false, false);
    }
    ap += 4;
    wp += 256;
  }
#pragma unroll
  for (int nt = 0; nt < 8; ++nt)
#pragma unroll
    for (int r = 0; r < 8; ++r)
      C[(size_t)(r0 + r) * HID + nt * 16 + l16] = acc[nt][r];
}

// One wave per edge; lane owns dims 4l..4l+3 (head = lane>>3).
__global__ void score_kernel(const float* __restrict__ q, const float* __restrict__ k,
                             const int* __restrict__ row, const int* __restrict__ col,
                             float* __restrict__ score, unsigned* __restrict__ m, int E) {
  int w = (blockIdx.x * blockDim.x + threadIdx.x) >> 5;
  if (w >= E) return;
  int lane = threadIdx.x & 31;
  int ci = col[w], ri = row[w];
  float4 qv = ((const float4*)(q + (size_t)ci * HID))[lane];
  float4 kv = ((const float4*)(k + (size_t)ri * HID))[lane];
  float p = qv.x * kv.x + qv.y * kv.y + qv.z * kv.z + qv.w * kv.w;
  p = head_red_sum(p);
  if ((lane & 7) == 0) {
    int hd = lane >> 3;
    float s = p * 0.17677669529663688f;  // 1/sqrt(32)
    score[(size_t)w * 4 + hd] = s;
    atomicMax(&m[(size_t)ci * 4 + hd], fkey(s));
  }
}

// One thread per edge: ex = exp(score - m[col]); z[col] += ex; store ex in place.
__global__ void ex_kernel(float* __restrict__ score, const int* __restrict__ col,
                          const unsigned* __restrict__ m, float* __restrict__ z, int E) {
  int e = blockIdx.x * blockDim.x + threadIdx.x;
  if (e >= E) return;
  int ci = col[e];
  float4 s = *(const float4*)(score + (size_t)e * 4);
  float e0 = expf(s.x - funkey(m[(size_t)ci * 4 + 0]));
  float e1 = expf(s.y - funkey(m[(size_t)ci * 4 + 1]));
  float e2 = expf(s.z - funkey(m[(size_t)ci * 4 + 2]));
  float e3 = expf(s.w - funkey(m[(size_t)ci * 4 + 3]));
  *(float4*)(score + (size_t)e * 4) = make_float4(e0, e1, e2, e3);
  atomic_fadd(&z[(size_t)ci * 4 + 0], e0);
  atomic_fadd(&z[(size_t)ci * 4 + 1], e1);
  atomic_fadd(&z[(size_t)ci * 4 + 2], e2);
  atomic_fadd(&z[(size_t)ci * 4 + 3], e3);
}

// One wave per edge: agg[col] += (ex/z[col]) * v[row]; lane owns dims 4l..4l+3.
__global__ void agg_kernel(const float* __restrict__ ex, const float* __restrict__ z,
                           const float* __restrict__ v, const int* __restrict__ row,
                           const int* __restrict__ col, float* __restrict__ agg, int E) {
  int w = (blockIdx.x * blockDim.x + threadIdx.x) >> 5;
  if (w >= E) return;
  int lane = threadIdx.x & 31;
  int ci = col[w], ri = row[w];
  int hd = lane >> 3;
  float al = ex[(size_t)w * 4 + hd] / z[(size_t)ci * 4 + hd];
  float4 vv = ((const float4*)(v + (size_t)ri * HID))[lane];
  float* ap = agg + (size_t)ci * HID + lane * 4;
  atomic_fadd(ap + 0, al * vv.x);
  atomic_fadd(ap + 1, al * vv.y);
  atomic_fadd(ap + 2, al * vv.z);
  atomic_fadd(ap + 3, al * vv.w);
}

// One wave per node: y = h + agg + skip; LayerNorm; ReLU; write back to h.
__global__ void ln_kernel(float* __restrict__ h, const float* __restrict__ agg,
                          const float* __restrict__ sk, const float* __restrict__ g,
                          const float* __restrict__ b, int N) {
  int w = (blockIdx.x * blockDim.x + threadIdx.x) >> 5;
  if (w >= N) return;
  int lane = threadIdx.x & 31;
  size_t base = (size_t)w * HID;
  float4 hv = ((const float4*)(h + base))[lane];
  float4 av = ((const float4*)(agg + base))[lane];
  float4 sv = ((const float4*)(sk + base))[lane];
  float y0 = hv.x + av.x + sv.x;
  float y1 = hv.y + av.y + sv.y;
  float y2 = hv.z + av.z + sv.z;
  float y3 = hv.w + av.w + sv.w;
  float mu = wave_red_sum(y0 + y1 + y2 + y3) * (1.0f / HID);
  float t0 = y0 - mu, t1 = y1 - mu, t2 = y2 - mu, t3 = y3 - mu;
  float var = wave_red_sum(t0 * t0 + t1 * t1 + t2 * t2 + t3 * t3) * (1.0f / HID);
  float rstd = rsqrtf(var + LN_EPS);
  float4 gv = ((const float4*)g)[lane];
  float4 bv = ((const float4*)b)[lane];
  float4 o;
  o.x = fmaxf(gv.x * (t0 * rstd) + bv.x, 0.0f);
  o.y = fmaxf(gv.y * (t1 * rstd) + bv.y, 0.0f);
  o.z = fmaxf(gv.z * (t2 * rstd) + bv.z, 0.0f);
  o.w = fmaxf(gv.w * (t3 * rstd) + bv.w, 0.0f);
  ((float4*)(h + base))[lane] = o;
}

// One wave per pair: out[p] = sigmoid(dot(h[src], h[dst])).
__global__ void logits_kernel(const float* __restrict__ h, const int* __restrict__ src,
                              const int* __restrict__ dst, float* __restrict__ out, int P) {
  int w = (blockIdx.x * blockDim.x + threadIdx.x) >> 5;
  if (w >= P) return;
  int lane = threadIdx.x & 31;
  float4 a = ((const float4*)(h + (size_t)src[w] * HID))[lane];
  float4 b = ((const float4*)(h + (size_t)dst[w] * HID))[lane];
  float p = a.x * b.x + a.y * b.y + a.z * b.z + a.w * b.w;
  p = wave_red_sum(p);
  if (lane == 0) out[w] = 1.0f / (1.0f + expf(-p));
}

static inline int cdiv(long long a, long long b) { return (int)((a + b - 1) / b); }

extern "C" void kernel_launch(void* const* d_in, const int* in_sizes, int n_in,
                              void* d_out, int out_size, void* d_ws, size_t ws_size,
                              hipStream_t stream) {
  const float* x       = (const float*)d_in[0];
  const float* rw_diag = (const float*)d_in[1];
  const int*   eidx    = (const int*)d_in[2];
  const int*   src     = (const int*)d_in[3];
  const int*   dst     = (const int*)d_in[4];
  const float* W_rwse  = (const float*)d_in[5];
  const float* b_rwse  = (const float*)d_in[6];
  const float* W_in    = (const float*)d_in[7];
  const float* b_in    = (const float*)d_in[8];
  const float* Wq      = (const float*)d_in[9];
  const float* bq      = (const float*)d_in[10];
  const float* Wk      = (const float*)d_in[11];
  const float* bk      = (const float*)d_in[12];
  const float* Wv      = (const float*)d_in[13];
  const float* bv      = (const float*)d_in[14];
  const float* Ws      = (const float*)d_in[15];
  const float* bs      = (const float*)d_in[16];
  const float* ln_g    = (const float*)d_in[17];
  const float* ln_b    = (const float*)d_in[18];
  float* out = (float*)d_out;

  const int N = in_sizes[0] / IN_C;
  const int E = in_sizes[2] / 2;
  const int P = in_sizes[3];
  const int* erow = eidx;       // message source j
  const int* ecol = eidx + E;   // message target i

  float* ws = (float*)d_ws;
  size_t nh = (size_t)N * HID;
  float* h    = ws;
  float* qb   = h + nh;
  float* kb   = qb + nh;
  float* vb   = kb + nh;
  float* aggb = vb + nh;
  float* sc   = aggb + nh;                     // E*4 floats; pe aliases this region
  unsigned* mbuf = (unsigned*)(sc + (size_t)E * 4);
  float* zbuf = (float*)(mbuf + (size_t)N * 4);
  float2* wpack = (float2*)(zbuf + (size_t)N * 4);  // up to 8192 float2 (64 KB)
  float* pe = sc;   // N*16 <= E*4

  const int BLK = 256;
  const int gemmBlocks = cdiv(N >> 4, 4);       // 4 waves per 128-thread block
  const int edgeWaveBlocks = cdiv((long long)E * 32, BLK);
  const int nodeWaveBlocks = cdiv((long long)N * 32, BLK);
  const int pack128Blocks = cdiv((128 >> 2) << 8, BLK);
  const int pack16Blocks  = cdiv((16 >> 2) << 8, BLK);

  // pe = rw_diag @ W_rwse + b_rwse
  pe_kernel<<<cdiv((long long)N * 16, BLK), BLK, 0, stream>>>(rw_diag, W_rwse, b_rwse, pe, N);

  // h = x @ W_in[:128] + b_in ; h += pe @ W_in[128:144]
  pack_w_kernel<<<pack128Blocks, BLK, 0, stream>>>(W_in, wpack, IN_C);
  gemm128_wmma<<<gemmBlocks, 128, 0, stream>>>(x, IN_C, IN_C, wpack, b_in, h, N, 0);
  pack_w_kernel<<<pack16Blocks, BLK, 0, stream>>>(W_in + (size_t)IN_C * HID, wpack, RWSE_DIM);
  gemm128_wmma<<<gemmBlocks, 128, 0, stream>>>(pe, RWSE_DIM, RWSE_DIM, wpack, nullptr, h, N, 1);

  for (int l = 0; l < NLAYERS; ++l) {
    const float* Wql = Wq + (size_t)l * HID * HID;
    const float* Wkl = Wk + (size_t)l * HID * HID;
    const float* Wvl = Wv + (size_t)l * HID * HID;
    const float* Wsl = Ws + (size_t)l * HID * HID;

    pack_w_kernel<<<pack128Blocks, BLK, 0, stream>>>(Wql, wpack, HID);
    gemm128_wmma<<<gemmBlocks, 128, 0, stream>>>(h, HID, HID, wpack, bq + l * HID, qb, N, 0);
    pack_w_kernel<<<pack128Blocks, BLK, 0, stream>>>(Wkl, wpack, HID);
    gemm128_wmma<<<gemmBlocks, 128, 0, stream>>>(h, HID, HID, wpack, bk + l * HID, kb, N, 0);
    pack_w_kernel<<<pack128Blocks, BLK, 0, stream>>>(Wvl, wpack, HID);
    gemm128_wmma<<<gemmBlocks, 128, 0, stream>>>(h, HID, HID, wpack, bv + l * HID, vb, N, 0);

    hipMemsetAsync(mbuf, 0, (size_t)N * 4 * sizeof(unsigned), stream);
    hipMemsetAsync(zbuf, 0, (size_t)N * 4 * sizeof(float), stream);
    hipMemsetAsync(aggb, 0, nh * sizeof(float), stream);

    score_kernel<<<edgeWaveBlocks, BLK, 0, stream>>>(qb, kb, erow, ecol, sc, mbuf, E);
    ex_kernel<<<cdiv(E, BLK), BLK, 0, stream>>>(sc, ecol, mbuf, zbuf, E);
    agg_kernel<<<edgeWaveBlocks, BLK, 0, stream>>>(sc, zbuf, vb, erow, ecol, aggb, E);

    // skip projection into qb (q no longer needed)
    pack_w_kernel<<<pack128Blocks, BLK, 0, stream>>>(Wsl, wpack, HID);
    gemm128_wmma<<<gemmBlocks, 128, 0, stream>>>(h, HID, HID, wpack, bs + l * HID, qb, N, 0);
    ln_kernel<<<nodeWaveBlocks, BLK, 0, stream>>>(h, aggb, qb, ln_g + l * HID,
                                                  ln_b + l * HID, N);
  }

  logits_kernel<<<cdiv((long long)P * 32, BLK), BLK, 0, stream>>>(h, src, dst, out, P);
  (void)n_in; (void)out_size; (void)ws_size;
}